// GraphAggrMLPv2_12541304504449
// MI455X (gfx1250) — compile-verified
//
#include <hip/hip_runtime.h>

typedef float v2f __attribute__((ext_vector_type(2)));
typedef float v8f __attribute__((ext_vector_type(8)));

#define NROWS 8192
#define MCOLS 128
#define KDZ   128
#define KDX   128
#define HEADS 4

// ---------------------------------------------------------------------------
// Kernel 1: xWb[m][h] = sum_k x[k][m] * W[KDZ+k][h] + b[h]   (128 x 4 floats)
// ---------------------------------------------------------------------------
__global__ void __launch_bounds__(128)
xw_bias_kernel(const float* __restrict__ x,
               const float* __restrict__ W,
               const float* __restrict__ b,
               float* __restrict__ xwb) {
  const int m = threadIdx.x;  // 0..127
  float a0 = b[0], a1 = b[1], a2 = b[2], a3 = b[3];
  for (int k = 0; k < KDX; ++k) {
    const float xv = x[k * MCOLS + m];
    const float* wr = W + (KDZ + k) * HEADS;
    a0 = fmaf(xv, wr[0], a0);
    a1 = fmaf(xv, wr[1], a1);
    a2 = fmaf(xv, wr[2], a2);
    a3 = fmaf(xv, wr[3], a3);
  }
  ((float4*)xwb)[m] = make_float4(a0, a1, a2, a3);
}

// ---------------------------------------------------------------------------
// Kernel 2: per wave, zW (16 rows x 4 heads) via 32x V_WMMA_F32_16X16X4_F32.
// B fragments (W chunks) are staged once per block into LDS with branch-free
// mask-multiply fills, so the K-loop is: global_load_b64 (z) + ds_load_b64 (B)
// + v_wmma per iteration. Then broadcast-add xWb from LDS and stream out
// coalesced float4 (b128) stores. 128 threads = 4 waves = 64 rows per block.
// ---------------------------------------------------------------------------
__global__ void __launch_bounds__(128)
fused_zw_bcast_kernel(const float* __restrict__ z,
                      const float* __restrict__ W,
                      const float* __restrict__ xwb,
                      float* __restrict__ out) {
  __shared__ float s_bf[32 * 64];            // 32 K-chunks x 64 floats (B frags)
  __shared__ float s_xwb[MCOLS * HEADS];     // 512 floats: xW + b, per m
  __shared__ float s_zw[4 * 16 * HEADS];     // 4 waves x 16 rows x 4 heads

  const int tid   = threadIdx.x;
  const int lane  = tid & 31;
  const int wave  = tid >> 5;      // 0..3
  const int nIdx  = lane & 15;     // A: row M / B: col N
  const int khalf = lane >> 4;     // 0: K={0,1}, 1: K={2,3}

  // Stage the broadcast term into LDS (one float4 per thread).
  ((float4*)s_xwb)[tid] = ((const float4*)xwb)[tid];

  // ---- Stage all 32 B fragments into LDS, branch-free.
  // Fragment layout for iteration i: lane l reads float2 at s_bf[i*64 + l*2].
  // Value pair = { W[(4i + 2*kh + 0)*4 + n], W[(4i + 2*kh + 1)*4 + n] } for
  // n = l&15 < 4, else 0 (mask-multiply; indices stay inside W's 1024 floats).
  {
    const int fl  = tid & 31;          // pseudo-lane this thread fills
    const int fi0 = tid >> 5;          // starting iteration (0..3)
    const int fn  = fl & 15;
    const int fkh = fl >> 4;
    const float msk = (fn < HEADS) ? 1.0f : 0.0f;
#pragma unroll
    for (int q = 0; q < 8; ++q) {
      const int i    = fi0 + 4 * q;          // iteration index 0..31
      const int krow = 4 * i + 2 * fkh;
      s_bf[i * 64 + fl * 2 + 0] = msk * W[(krow + 0) * HEADS + fn];
      s_bf[i * 64 + fl * 2 + 1] = msk * W[(krow + 1) * HEADS + fn];
    }
  }
  __syncthreads();

  // ---- zW = z @ W[:128] for this wave's 16 rows, fp32 WMMA, K-chunks of 4.
  const int n0 = blockIdx.x * 64 + wave * 16;
  const float* zrow = z + (size_t)(n0 + nIdx) * KDZ + 2 * khalf;

  v8f c = {0.f, 0.f, 0.f, 0.f, 0.f, 0.f, 0.f, 0.f};
#pragma unroll
  for (int i = 0; i < KDZ / 4; ++i) {
    // A fragment: lanes 0-15 hold z[n, 4i..4i+1], lanes 16-31 hold z[n, 4i+2..4i+3]
    v2f a  = *(const v2f*)(zrow + 4 * i);
    // B fragment from LDS (ds_load_b64, conflict-free, shared by all waves)
    v2f bb = *(const v2f*)(s_bf + i * 64 + lane * 2);
    // 8 args: (neg_a, A, neg_b, B, c_mod, C, reuse_a, reuse_b)
    c = __builtin_amdgcn_wmma_f32_16x16x4_f32(false, a, false, bb,
                                              (short)0, c, false, false);
  }

  // ---- Scatter the 4 valid accumulator columns (N = head) into LDS.
  // C layout: VGPR r -> lanes 0-15: (M=r, N=lane); lanes 16-31: (M=r+8, N=lane-16)
  if (nIdx < HEADS) {
    const int base = wave * 64 + khalf * (8 * HEADS) + nIdx;
#pragma unroll
    for (int r = 0; r < 8; ++r) {
      s_zw[base + r * HEADS] = c[r];
    }
  }
  __syncthreads();

  // ---- Broadcast add + relu; 64 coalesced b128 stores per wave.
#pragma unroll 4
  for (int r = 0; r < 16; ++r) {
    const float4 zw = *(const float4*)(s_zw + wave * 64 + r * HEADS);
    const size_t rowoff = (size_t)(n0 + r) * (MCOLS * HEADS);
#pragma unroll
    for (int mb = 0; mb < 4; ++mb) {
      const int m = mb * 32 + lane;
      const float4 xw = *(const float4*)(s_xwb + m * HEADS);
      float4 o;
      o.x = fmaxf(zw.x + xw.x, 0.f);
      o.y = fmaxf(zw.y + xw.y, 0.f);
      o.z = fmaxf(zw.z + xw.z, 0.f);
      o.w = fmaxf(zw.w + xw.w, 0.f);
      *(float4*)(out + rowoff + (size_t)m * HEADS) = o;
    }
  }
}

// ---------------------------------------------------------------------------
extern "C" void kernel_launch(void* const* d_in, const int* in_sizes, int n_in,
                              void* d_out, int out_size, void* d_ws, size_t ws_size,
                              hipStream_t stream) {
  (void)in_sizes; (void)n_in; (void)out_size; (void)ws_size;
  const float* z = (const float*)d_in[0];   // [8192, 128]
  const float* x = (const float*)d_in[1];   // [128, 128]
  const float* W = (const float*)d_in[2];   // [256, 4]
  const float* b = (const float*)d_in[3];   // [4]
  float* out = (float*)d_out;               // [8192, 128, 4]
  float* xwb = (float*)d_ws;                // 512 floats scratch

  xw_bias_kernel<<<1, 128, 0, stream>>>(x, W, b, xwb);
  fused_zw_bcast_kernel<<<NROWS / 64, 128, 0, stream>>>(z, W, xwb, out);
}